// DifferenceOfGaussiansFFT_50062138802716
// MI455X (gfx1250) — compile-verified
//
#include <hip/hip_runtime.h>
#include <math.h>

// ---------------------------------------------------------------------------
// Difference-of-Gaussians (FFT reference) as separable fp32 Gaussian blurs on
// the CDNA5 matrix core (V_WMMA_F32_16X16X4_F32), then DoG + 3x3x3 max/mask.
//
// Branch-free WMMA inner loops: zero-padded LDS strips + offset weight table
// make every ds_load unconditional and keep EXEC all-ones at each WMMA.
//
// ws layout (floats):
//   [0 .. 4224)                  : 33 levels x 128 normalized 1-D taps
//   [4352 .. 4352+2*33*512^2)    : tmp (h-blur result; later aliased as dog)
//   [G_OFF .. G_OFF+2*33*512^2)  : g (blurred pyramid)          (~138.5 MB)
// ---------------------------------------------------------------------------

typedef __attribute__((ext_vector_type(2))) float v2f;
typedef __attribute__((ext_vector_type(8))) float v8f;

#define IMG     512
#define LEVELS  33
#define NB      2
#define PLANE   (IMG * IMG)          // 2^18
#define THRESH  1.0e-3f
#define NEGBIG  (-3.402823466e38f)

#define WT_OFF   0
#define TMP_OFF  4352
#define TMP_SZ   ((size_t)NB * LEVELS * PLANE)
#define G_OFF    (TMP_OFF + TMP_SZ)

#define SCOLS    236                 // strip extent: 128 + 2*51 = 230 max, +pad
#define WPAD     16                  // weight table left zero-pad
#define WTAB     144                 // 16 + 128

// --------------------------------------------------------------------------
// 1) Normalized separable taps: w[t] = exp(-((t-r)/s)^2/2) / sum, zero-padded
// --------------------------------------------------------------------------
__global__ void dog_weights(const float* __restrict__ sig, float* __restrict__ wt) {
    const int lvl = blockIdx.x;
    const float s = sig[lvl];
    const int r = (int)(5.0f * s + 0.5f);
    const int width = 2 * r + 1;          // <= 103
    const int t = threadIdx.x;            // 128 threads
    float v = 0.f;
    if (t < width) {
        float d = ((float)(t - r)) / s;
        v = __expf(-0.5f * d * d);
    }
    __shared__ float red[128];
    red[t] = v;
    __syncthreads();
    for (int off = 64; off > 0; off >>= 1) {
        if (t < off) red[t] += red[t + off];
        __syncthreads();
    }
    wt[lvl * 128 + t] = (t < width) ? (v / red[0]) : 0.f;
}

// --------------------------------------------------------------------------
// 2) Horizontal blur. Wave = one 16x16 tile; block = 16 rows x 128 cols.
//    O[m,n] = sum_c I[y0+m,c] * w[c-(x0+n)+r];  A = data strip, B = band.
// --------------------------------------------------------------------------
__global__ __launch_bounds__(256) void dog_hblur(const float* __restrict__ img,
                                                 const float* __restrict__ sig,
                                                 const float* __restrict__ wt,
                                                 float* __restrict__ tmp) {
    const int lvl = blockIdx.y;
    const int b   = blockIdx.z;
    const int xg  = blockIdx.x & 3;       // 4 col-groups of 128
    const int ys  = blockIdx.x >> 2;      // 32 row strips of 16
    const int r   = (int)(5.0f * sig[lvl] + 0.5f);
    const int tid = threadIdx.x;

    __shared__ float wsh[WTAB];           // [0,16)=0, taps at +16, 0 beyond
    __shared__ float strip[16][SCOLS];    // fully zero-padded image strip

    if (tid < WTAB)
        wsh[tid] = (tid >= WPAD) ? wt[lvl * 128 + (tid - WPAD)] : 0.f;

    const int xbase = xg * 128;
    const int y0    = ys * 16;
    const int g0    = xbase - r;          // global col of strip col 0
    const int nc    = 128 + 2 * r;        // valid strip cols (<=230)
    const float* ib = img + (size_t)b * PLANE;
    if (tid < SCOLS) {
        const int gc  = g0 + tid;
        const bool ok = (tid < nc) & (gc >= 0) & (gc < IMG);
        const int gcc = min(max(gc, 0), IMG - 1);
#pragma unroll
        for (int row = 0; row < 16; ++row) {
            const float v = ib[(y0 + row) * IMG + gcc];   // clamped, always valid
            strip[row][tid] = ok ? v : 0.f;
        }
    }
    __syncthreads();

    const int lane = tid & 31;
    const int wv   = tid >> 5;            // 8 waves -> 8 x-tiles
    const int n    = lane & 15;           // N (== M for A reads) per ISA layout
    const int j0   = (lane >> 4) << 1;    // K base: 0 (lanes 0-15) / 2 (16-31)

    v8f acc = {};
    const int nk = (2 * r + 16 + 3) >> 2;
    const int abase = wv * 16 + j0;             // strip col for chunk 0
    const int bbase = WPAD + j0 - n;            // weight index for chunk 0
    for (int kk = 0; kk < nk; ++kk) {
        const int sI = kk << 2;
        v2f A, B;
        A.x = strip[n][abase + sI];             // unconditional LDS reads
        A.y = strip[n][abase + sI + 1];
        B.x = wsh[bbase + sI];
        B.y = wsh[bbase + sI + 1];
        acc = __builtin_amdgcn_wmma_f32_16x16x4_f32(false, A, false, B,
                                                    (short)0, acc, false, false);
    }

    float* ob = tmp + ((size_t)(b * LEVELS + lvl)) * PLANE;
    const int x     = xbase + wv * 16 + n;
    const int rbase = y0 + ((lane < 16) ? 0 : 8);  // C layout: lanes 16-31 = M+8
#pragma unroll
    for (int v = 0; v < 8; ++v)
        ob[(rbase + v) * IMG + x] = acc[v];
}

// --------------------------------------------------------------------------
// 3) Vertical blur. Block = 128 rows x 16 cols (8 waves stacked in y).
//    O[m,n] = sum_c w[c-(y0+m)+r] * T[c,x0+n];  A = band, B = data strip.
// --------------------------------------------------------------------------
__global__ __launch_bounds__(256) void dog_vblur(const float* __restrict__ tmp,
                                                 const float* __restrict__ sig,
                                                 const float* __restrict__ wt,
                                                 float* __restrict__ g) {
    const int lvl = blockIdx.y;
    const int b   = blockIdx.z;
    const int tx  = blockIdx.x & 31;      // 32 x-tiles
    const int yg  = blockIdx.x >> 5;      // 4 y-groups of 128 rows
    const int r   = (int)(5.0f * sig[lvl] + 0.5f);
    const int tid = threadIdx.x;

    __shared__ float wsh[WTAB];
    __shared__ float strip[SCOLS][16];    // rows [ybase-r, ...) x 16 cols

    if (tid < WTAB)
        wsh[tid] = (tid >= WPAD) ? wt[lvl * 128 + (tid - WPAD)] : 0.f;

    const int ybase = yg * 128;
    const int x0    = tx * 16;
    const int nrows = 128 + 2 * r;        // valid strip rows (<=230)
    const float* ib = tmp + ((size_t)(b * LEVELS + lvl)) * PLANE;
    for (int i = tid; i < SCOLS * 16; i += 256) {
        const int lr  = i >> 4;
        const int col = i & 15;
        const int gr  = ybase - r + lr;
        const bool ok = (lr < nrows) & (gr >= 0) & (gr < IMG);
        const int grc = min(max(gr, 0), IMG - 1);
        const float v = ib[grc * IMG + x0 + col];
        strip[lr][col] = ok ? v : 0.f;
    }
    __syncthreads();

    const int lane = tid & 31;
    const int wv   = tid >> 5;            // 8 waves -> 8 y-tiles
    const int y0   = ybase + wv * 16;
    const int n    = lane & 15;
    const int j0   = (lane >> 4) << 1;

    v8f acc = {};
    const int nk = (2 * r + 16 + 3) >> 2;
    const int abase = WPAD + j0 - n;            // weight index (m == n role)
    const int bbase = wv * 16 + j0;             // strip row for chunk 0
    for (int kk = 0; kk < nk; ++kk) {
        const int sI = kk << 2;
        v2f A, B;
        A.x = wsh[abase + sI];
        A.y = wsh[abase + sI + 1];
        B.x = strip[bbase + sI][n];
        B.y = strip[bbase + sI + 1][n];
        acc = __builtin_amdgcn_wmma_f32_16x16x4_f32(false, A, false, B,
                                                    (short)0, acc, false, false);
    }

    float* ob = g + ((size_t)(b * LEVELS + lvl)) * PLANE;
    const int rbase = y0 + ((lane < 16) ? 0 : 8);
#pragma unroll
    for (int v = 0; v < 8; ++v)
        ob[(rbase + v) * IMG + x0 + n] = acc[v];
}

// --------------------------------------------------------------------------
// 4) dog[b,i] = (g[b,i] - g[b,i+1]) * sigma[i]
// --------------------------------------------------------------------------
__global__ void dog_diff(const float* __restrict__ g, const float* __restrict__ sig,
                         float* __restrict__ dog) {
    const size_t i = (size_t)blockIdx.x * blockDim.x + threadIdx.x;
    const size_t N = (size_t)NB * 32 * PLANE;
    if (i >= N) return;
    const int b   = (int)(i >> 23);
    const int lvl = (int)((i >> 18) & 31);
    const size_t p = i & (PLANE - 1);
    const float* g0 = g + ((size_t)(b * LEVELS + lvl)) * PLANE;
    dog[i] = (g0[p] - g0[PLANE + p]) * sig[lvl];
}

// --------------------------------------------------------------------------
// 5) 3x3x3 local maxima (-inf padding == exclude OOB) + mask; branchless.
//    d_out = [mask (0/1 float), local_maxima], each NB*32*512*512.
// --------------------------------------------------------------------------
__global__ void dog_maxmask(const float* __restrict__ dog, float* __restrict__ out) {
    const size_t i = (size_t)blockIdx.x * blockDim.x + threadIdx.x;
    const size_t N = (size_t)NB * 32 * PLANE;
    if (i >= N) return;
    const int x   = (int)(i & 511);
    const int y   = (int)((i >> 9) & 511);
    const int lvl = (int)((i >> 18) & 31);
    const int b   = (int)(i >> 23);
    const float c = dog[i];
    const float* baseb = dog + ((size_t)b << 23);
    float lm = NEGBIG;                 // window always contains center
#pragma unroll
    for (int dl = -1; dl <= 1; ++dl) {
        const int l2  = lvl + dl;
        const bool lok = (l2 >= 0) & (l2 < 32);
        const int l2c = min(max(l2, 0), 31);
#pragma unroll
        for (int dy = -1; dy <= 1; ++dy) {
            const int y2  = y + dy;
            const bool yok = (y2 >= 0) & (y2 < IMG);
            const int y2c = min(max(y2, 0), IMG - 1);
#pragma unroll
            for (int dx = -1; dx <= 1; ++dx) {
                const int x2  = x + dx;
                const bool xok = (x2 >= 0) & (x2 < IMG);
                const int x2c = min(max(x2, 0), IMG - 1);
                const float v = baseb[((size_t)l2c << 18) + ((size_t)y2c << 9) + x2c];
                lm = fmaxf(lm, (lok & yok & xok) ? v : NEGBIG);
            }
        }
    }
    out[i]     = (lm == c && c > THRESH) ? 1.f : 0.f;
    out[N + i] = lm;
}

// --------------------------------------------------------------------------
extern "C" void kernel_launch(void* const* d_in, const int* in_sizes, int n_in,
                              void* d_out, int out_size, void* d_ws, size_t ws_size,
                              hipStream_t stream) {
    (void)in_sizes; (void)n_in; (void)out_size; (void)ws_size;
    const float* img = (const float*)d_in[0];
    // d_in[1] = padded FFT kernels: redundant (separable), rebuilt from sigmas.
    const float* sig = (const float*)d_in[2];

    float* ws  = (float*)d_ws;
    float* wt  = ws + WT_OFF;
    float* tmp = ws + TMP_OFF;
    float* g   = ws + G_OFF;
    float* dog = ws + TMP_OFF;            // alias: tmp dead after vblur
    float* out = (float*)d_out;

    dog_weights<<<LEVELS, 128, 0, stream>>>(sig, wt);
    dog_hblur<<<dim3(128, LEVELS, NB), 256, 0, stream>>>(img, sig, wt, tmp);
    dog_vblur<<<dim3(128, LEVELS, NB), 256, 0, stream>>>(tmp, sig, wt, g);

    const size_t N = (size_t)NB * 32 * PLANE;
    dog_diff<<<(unsigned)((N + 255) / 256), 256, 0, stream>>>(g, sig, dog);
    dog_maxmask<<<(unsigned)((N + 255) / 256), 256, 0, stream>>>(dog, out);
}